// HeteroLightGCN_68745246540300
// MI455X (gfx1250) — compile-verified
//
#include <hip/hip_runtime.h>
#include <hip/hip_bf16.h>

#define DIM 64           // embedding dim (reference: 64)
#define DIM4 (DIM / 4)   // float4 per row
#define WAVES_PER_BLOCK 8

#if __has_builtin(__builtin_amdgcn_global_load_async_to_lds_b32) && \
    __has_builtin(__builtin_amdgcn_s_wait_asynccnt)
#define USE_ASYNC_LDS 1
#else
#define USE_ASYNC_LDS 0
#endif

#if USE_ASYNC_LDS
// Param 1: non-const int* in address_space(1); param 2: LDS dest; imm offset, imm cpol.
__device__ __forceinline__ void HLGCN_async_b32(const int* gsrc, int* lds_dst) {
  __builtin_amdgcn_global_load_async_to_lds_b32(
      (__attribute__((address_space(1))) int*)(int*)gsrc,
      (__attribute__((address_space(3))) int*)lds_dst, 0, 0);
}
#endif

// ---------------------------------------------------------------- degree count
__global__ void HLGCN_deg_kernel(const int* __restrict__ esrc,
                                 const int* __restrict__ edst,
                                 int* __restrict__ deg_u,
                                 int* __restrict__ deg_m, int n) {
  int stride = gridDim.x * blockDim.x;
  for (int i = blockIdx.x * blockDim.x + threadIdx.x; i < n; i += stride) {
    atomicAdd(&deg_u[esrc[i]], 1);
    atomicAdd(&deg_m[edst[i]], 1);
  }
}

// ---------------------------------------------------------------- deg -> deg^-1/2
__global__ void HLGCN_dinv_kernel(const int* __restrict__ deg,
                                  float* __restrict__ dinv, int n) {
  int i = blockIdx.x * blockDim.x + threadIdx.x;
  if (i < n) {
    int d = deg[i];
    dinv[i] = (d > 0) ? rsqrtf((float)d) : 0.0f;
  }
}

// ---------------------------------------------------------------- out = 0.5 * emb[ids]
__global__ void HLGCN_init_kernel(const float4* __restrict__ uemb,
                                  const float4* __restrict__ memb,
                                  const int* __restrict__ uid,
                                  const int* __restrict__ mid,
                                  float4* __restrict__ out, int nu, int nm) {
  int total = (nu + nm) * DIM4;
  int i = blockIdx.x * blockDim.x + threadIdx.x;
  if (i >= total) return;
  int row = i / DIM4;
  int c = i - row * DIM4;
  float4 v;
  if (row < nu)
    v = uemb[(size_t)uid[row] * DIM4 + c];
  else
    v = memb[(size_t)mid[row - nu] * DIM4 + c];
  v.x *= 0.5f; v.y *= 0.5f; v.z *= 0.5f; v.w *= 0.5f;
  out[i] = v;
}

// ---------------------------------------------------------------- edge scatter
// One wave32 owns a group of 32 edges. Lanes pre-gather the 32 per-edge node ids,
// embedding-row ids and scales; the inner loop broadcasts each edge's scalars to
// SGPRs via v_readlane (no SMEM / KMcnt on the critical path), then the 32 lanes
// move the 64-dim row as float2 (one coalesced 256B line per gather) and issue
// no-return global_atomic_add_f32 into both outputs.
__device__ __forceinline__ void HLGCN_process_group(
    int cnt, int lane, int s_i, int d_i,
    const float* __restrict__ uemb, const float* __restrict__ memb,
    const int* __restrict__ uid, const int* __restrict__ mid,
    const float* __restrict__ dinv_u, const float* __restrict__ dinv_m,
    float* __restrict__ out_u, float* __restrict__ out_m) {
  float nrm = 0.5f * dinv_u[s_i] * dinv_m[d_i];
  int nrm_b = __float_as_int(nrm);
  int su_l = uid[s_i];  // per-lane vector gather, hoisted out of the k-loop
  int dm_l = mid[d_i];
#pragma unroll 4
  for (int k = 0; k < cnt; ++k) {
    int ss = __builtin_amdgcn_readlane(s_i, k);   // SGPR: output row (user)
    int dd = __builtin_amdgcn_readlane(d_i, k);   // SGPR: output row (movie)
    int su = __builtin_amdgcn_readlane(su_l, k);  // SGPR: user emb row
    int dm = __builtin_amdgcn_readlane(dm_l, k);  // SGPR: movie emb row
    float sc = __int_as_float(__builtin_amdgcn_readlane(nrm_b, k));
    const float2 xv = *(const float2*)(uemb + (size_t)su * DIM + lane * 2);
    const float2 yv = *(const float2*)(memb + (size_t)dm * DIM + lane * 2);
    float* om = out_m + (size_t)dd * DIM + lane * 2;
    float* ou = out_u + (size_t)ss * DIM + lane * 2;
    unsafeAtomicAdd(om,     sc * xv.x);   // x2y: movies += norm * x[src]
    unsafeAtomicAdd(om + 1, sc * xv.y);
    unsafeAtomicAdd(ou,     sc * yv.x);   // y2x: users  += norm * y[dst]
    unsafeAtomicAdd(ou + 1, sc * yv.y);
  }
}

__global__ __launch_bounds__(WAVES_PER_BLOCK * 32) void HLGCN_edge_kernel(
    const float* __restrict__ uemb, const float* __restrict__ memb,
    const int* __restrict__ uid, const int* __restrict__ mid,
    const int* __restrict__ esrc, const int* __restrict__ edst,
    const float* __restrict__ dinv_u, const float* __restrict__ dinv_m,
    float* __restrict__ out_u, float* __restrict__ out_m, int n) {
  const int lane = threadIdx.x & 31;
  const int wloc = threadIdx.x >> 5;
  const int wave = blockIdx.x * WAVES_PER_BLOCK + wloc;
  const int nwav = gridDim.x * WAVES_PER_BLOCK;
  const int ngroups = (n + 31) >> 5;

#if USE_ASYNC_LDS
  // Double-buffered async prefetch of edge-index blocks: CDNA5 async-to-LDS path.
  __shared__ int lds_idx[WAVES_PER_BLOCK][2][2][32];  // [wave][buf][src|dst][lane]
  int buf = 0;
  {
    int g0 = wave;
    if (g0 < ngroups) {
      int e = min(g0 * 32 + lane, n - 1);
      HLGCN_async_b32(esrc + e, &lds_idx[wloc][0][0][lane]);
      HLGCN_async_b32(edst + e, &lds_idx[wloc][0][1][lane]);
    }
  }
  for (int g = wave; g < ngroups; g += nwav) {
    int gn = g + nwav;
    bool hn = gn < ngroups;
    if (hn) {
      int e = min(gn * 32 + lane, n - 1);
      HLGCN_async_b32(esrc + e, &lds_idx[wloc][buf ^ 1][0][lane]);
      HLGCN_async_b32(edst + e, &lds_idx[wloc][buf ^ 1][1][lane]);
      __builtin_amdgcn_s_wait_asynccnt(2);  // current group's 2 loads complete
    } else {
      __builtin_amdgcn_s_wait_asynccnt(0);
    }
    int s_i = lds_idx[wloc][buf][0][lane];
    int d_i = lds_idx[wloc][buf][1][lane];
    int cnt = min(32, n - (g << 5));
    HLGCN_process_group(cnt, lane, s_i, d_i, uemb, memb, uid, mid,
                        dinv_u, dinv_m, out_u, out_m);
    buf ^= 1;
  }
#else
  for (int g = wave; g < ngroups; g += nwav) {
    int e = min(g * 32 + lane, n - 1);
    int s_i = esrc[e];
    int d_i = edst[e];
    int cnt = min(32, n - (g << 5));
    HLGCN_process_group(cnt, lane, s_i, d_i, uemb, memb, uid, mid,
                        dinv_u, dinv_m, out_u, out_m);
  }
#endif
}

// ---------------------------------------------------------------- launch
extern "C" void kernel_launch(void* const* d_in, const int* in_sizes, int n_in,
                              void* d_out, int out_size, void* d_ws, size_t ws_size,
                              hipStream_t stream) {
  const float* uemb = (const float*)d_in[0];
  const float* memb = (const float*)d_in[1];
  const int* uid = (const int*)d_in[2];
  const int* mid = (const int*)d_in[3];
  const int* esrc = (const int*)d_in[4];
  const int* edst = (const int*)d_in[5];

  const int nu = in_sizes[2];
  const int nm = in_sizes[3];
  const int ne = in_sizes[4];

  float* out_u = (float*)d_out;
  float* out_m = out_u + (size_t)nu * DIM;

  // workspace: deg (int[nu+nm]) | dinv (float[nu+nm])
  int* deg = (int*)d_ws;
  float* dinv = (float*)(deg + (size_t)(nu + nm));
  float* dinv_u = dinv;
  float* dinv_m = dinv + nu;

  (void)hipMemsetAsync(deg, 0, sizeof(int) * (size_t)(nu + nm), stream);

  HLGCN_deg_kernel<<<4096, 256, 0, stream>>>(esrc, edst, deg, deg + nu, ne);

  int tot = nu + nm;
  HLGCN_dinv_kernel<<<(tot + 255) / 256, 256, 0, stream>>>(deg, dinv, tot);

  int tot4 = tot * DIM4;
  HLGCN_init_kernel<<<(tot4 + 255) / 256, 256, 0, stream>>>(
      (const float4*)uemb, (const float4*)memb, uid, mid, (float4*)d_out, nu, nm);

  HLGCN_edge_kernel<<<4096, WAVES_PER_BLOCK * 32, 0, stream>>>(
      uemb, memb, uid, mid, esrc, edst, dinv_u, dinv_m, out_u, out_m, ne);
}